// SINRnet_6777458393534
// MI455X (gfx1250) — compile-verified
//
#include <hip/hip_runtime.h>
#include <math.h>

typedef float v2f __attribute__((ext_vector_type(2)));
typedef float v8f __attribute__((ext_vector_type(8)));

#define NB 64          // batch
#define NN 64          // spatial N
#define C4 128         // conv hidden channels
#define KPAD 132       // 129 in-channels padded to multiple of 4

// ---- workspace layout (in floats) ----
#define Y_OFF      0
#define Y_SIZE     (NB * C4 * NN * NN)        // 33,554,432 floats (128 MB)
#define S_PER_B    8256                        // C1[32][64],R2,R3,C3 @2048 each, T3[32] @8192
#define S_OFF      (Y_OFF + Y_SIZE)
#define S_SIZE     (NB * S_PER_B)
#define WT_OFF     (S_OFF + S_SIZE)
#define WT_SIZE    (6 * KPAD * C4)
#define BRA_OFF    (WT_OFF + WT_SIZE)
#define BRB_OFF    (BRA_OFF + NB * NN)

// ------------------------------------------------------------------
// Transpose + pad the 6 hidden conv weights (128x129 -> 132x128 K-major)
__global__ void k_wt(const float* w0, const float* w1, const float* w2,
                     const float* w3, const float* w4, const float* w5,
                     float* __restrict__ wt) {
    const float* wsel[6] = {w0, w1, w2, w3, w4, w5};
    const float* w = wsel[blockIdx.x];
    float* o = wt + (size_t)blockIdx.x * (KPAD * C4);
    for (int idx = threadIdx.x; idx < KPAD * C4; idx += blockDim.x) {
        int k = idx >> 7, m = idx & 127;
        o[idx] = (k < 129) ? w[m * 129 + k] : 0.0f;
    }
}

// ------------------------------------------------------------------
// Layer-1 conv: y[b,c,i,j] = relu(w1[c]*channel[b,i,j] + b1[c])
__global__ void k_conv1(const float* __restrict__ ch, const float* __restrict__ w1,
                        const float* __restrict__ b1, float* __restrict__ y) {
    int b = blockIdx.x >> 7;
    int c = blockIdx.x & 127;
    int j = threadIdx.x;                 // 64 threads
    float wv = w1[c], bv = b1[c];
    const float* chp = ch + b * NN * NN;
    float* yp = y + ((size_t)(b * C4 + c)) * NN * NN;
    for (int i = 0; i < NN; ++i) {
        float v = wv * chp[i * NN + j] + bv;
        yp[i * NN + j] = v > 0.0f ? v : 0.0f;
    }
}

// ------------------------------------------------------------------
// Row/col/total sums of y channel groups (ex-matmul reduction).
// grid = NB*3 : g=0 -> col sums of ch[32,64)   (for c1 = ex@t)
//               g=1 -> row sums of ch[64,96)   (for c2 = t@ex)
//               g=2 -> row+col+total, ch[96,128) (for c3 = ex@t@ex)
__global__ void k_sums(const float* __restrict__ y, float* __restrict__ sums) {
    __shared__ float plane[64 * 65];     // pitch 65: conflict-free row & col reads
    __shared__ float rsum[64];
    int b = blockIdx.x / 3, g = blockIdx.x % 3;
    int tid = threadIdx.x;
    float* S = sums + (size_t)b * S_PER_B;
    for (int cc = 0; cc < 32; ++cc) {
        int chn = 32 + g * 32 + cc;
        const float* yp = y + ((size_t)(b * C4 + chn)) * NN * NN;
        for (int idx = tid; idx < 4096; idx += 256)
            plane[(idx >> 6) * 65 + (idx & 63)] = yp[idx];
        __syncthreads();
        if (g == 0) {
            if (tid < 64) { float s = 0;
                for (int i2 = 0; i2 < 64; ++i2) s += plane[i2 * 65 + tid];
                S[cc * 64 + tid] = s; }
        } else if (g == 1) {
            if (tid < 64) { float s = 0;
                for (int j2 = 0; j2 < 64; ++j2) s += plane[tid * 65 + j2];
                S[2048 + cc * 64 + tid] = s; }
        } else {
            if (tid < 64) { float s = 0;
                for (int j2 = 0; j2 < 64; ++j2) s += plane[tid * 65 + j2];
                S[4096 + cc * 64 + tid] = s; rsum[tid] = s; }
            else if (tid < 128) { int j3 = tid - 64; float s = 0;
                for (int i2 = 0; i2 < 64; ++i2) s += plane[i2 * 65 + j3];
                S[6144 + cc * 64 + j3] = s; }
            __syncthreads();
            if (tid == 0) { float t = 0;
                for (int i2 = 0; i2 < 64; ++i2) t += rsum[i2];
                S[8192 + cc] = t; }
        }
        __syncthreads();
    }
}

// ------------------------------------------------------------------
// Fused post+conv+relu for layers 2..4, in place on y.
// Block = (b, image row i); 8 waves; wave w -> out-channel strip [16w,16w+16).
// x tile (132 x 64) rebuilt in LDS from y + sums + channel, then fp32 WMMA GEMM.
__global__ void k_conv(const float* __restrict__ yin, const float* __restrict__ ch,
                       const float* __restrict__ sums, const float* __restrict__ wt,
                       const float* __restrict__ bias, float* __restrict__ yout) {
    __shared__ float xs[KPAD * NN];      // 33 KB
    int b = blockIdx.x >> 6;
    int i = blockIdx.x & 63;
    int tid = threadIdx.x;
    int j = tid & 63, kq = tid >> 6;

    const float* S  = sums + (size_t)b * S_PER_B;
    const float* yb = yin + (size_t)b * C4 * NN * NN;
    const float inv1 = 1.0f / 63.0f, inv2 = 1.0f / 3969.0f;

    // ---- build x tile: [channel, c0, c1, c2, c3] at row i ----
    for (int k = kq; k < KPAD; k += 4) {
        float v;
        if (k == 0) {
            v = ch[b * NN * NN + i * NN + j];
        } else if (k <= 32) {            // c0 = y
            int c = k - 1;
            v = yb[(c * NN + i) * NN + j];
        } else if (k <= 64) {            // c1 = (Col_j - y)/63
            int c = k - 1;
            v = (S[(c - 32) * 64 + j] - yb[(c * NN + i) * NN + j]) * inv1;
        } else if (k <= 96) {            // c2 = (Row_i - y)/63
            int c = k - 1;
            v = (S[2048 + (c - 64) * 64 + i] - yb[(c * NN + i) * NN + j]) * inv1;
        } else if (k <= 128) {           // c3 = (T - R_i - C_j + y)/3969
            int c = k - 1; int gg = c - 96;
            v = (S[8192 + gg] - S[4096 + gg * 64 + i] - S[6144 + gg * 64 + j]
                 + yb[(c * NN + i) * NN + j]) * inv2;
        } else {
            v = 0.0f;                    // K padding 129..131
        }
        xs[k * NN + j] = v;
    }
    __syncthreads();

    // ---- GEMM: 16x16x4 fp32 WMMA ----
    int wv   = tid >> 5;                 // wave 0..7
    int lane = tid & 31;
    int half = lane >> 4, row = lane & 15;
    int mbase = wv * 16;

    v8f zero = {};
    v8f acc[4] = {zero, zero, zero, zero};

    for (int kk = 0; kk < 33; ++kk) {
        int kb = kk * 4 + 2 * half;      // A vgpr0 holds K=0 (lanes<16) / K=2 (lanes>=16)
        v2f af;
        af[0] = wt[kb * C4 + mbase + row];
        af[1] = wt[(kb + 1) * C4 + mbase + row];
#pragma unroll
        for (int pt = 0; pt < 4; ++pt) {
            v2f bf;
            bf[0] = xs[kb * NN + pt * 16 + row];
            bf[1] = xs[(kb + 1) * NN + pt * 16 + row];
            acc[pt] = __builtin_amdgcn_wmma_f32_16x16x4_f32(
                false, af, false, bf, (short)0, acc[pt], false, false);
        }
    }

    // ---- bias + relu + store (in place) ----
    float bl[8];
#pragma unroll
    for (int r = 0; r < 8; ++r) bl[r] = bias[mbase + r + 8 * half];
    float* yo = yout + (size_t)b * C4 * NN * NN;
#pragma unroll
    for (int pt = 0; pt < 4; ++pt) {
        int jj = pt * 16 + row;
#pragma unroll
        for (int r = 0; r < 8; ++r) {
            int oc = mbase + r + 8 * half;     // D vgpr r: M=r (lanes<16) / M=r+8
            float v = acc[pt][r] + bl[r];
            yo[(oc * NN + i) * NN + jj] = v > 0.0f ? v : 0.0f;
        }
    }
}

// ------------------------------------------------------------------
// Layer-5 conv evaluated only on the diagonal + post fusion.
__global__ void k_diag(const float* __restrict__ y, const float* __restrict__ ch,
                       const float* __restrict__ sums, const float* __restrict__ w5,
                       const float* __restrict__ b5, float* __restrict__ outb) {
    int b = blockIdx.x, i = threadIdx.x;     // 64 threads
    const float* S  = sums + (size_t)b * S_PER_B;
    const float* yb = y + (size_t)b * C4 * NN * NN;
    const float inv1 = 1.0f / 63.0f, inv2 = 1.0f / 3969.0f;
    float acc = b5[0] + w5[0] * ch[b * NN * NN + i * NN + i];
    for (int c = 0; c < 32; ++c)
        acc += w5[1 + c] * yb[(c * NN + i) * NN + i];
    for (int c = 32; c < 64; ++c)
        acc += w5[1 + c] * ((S[(c - 32) * 64 + i] - yb[(c * NN + i) * NN + i]) * inv1);
    for (int c = 64; c < 96; ++c)
        acc += w5[1 + c] * ((S[2048 + (c - 64) * 64 + i] - yb[(c * NN + i) * NN + i]) * inv1);
    for (int c = 96; c < 128; ++c) { int g = c - 96;
        acc += w5[1 + c] * ((S[8192 + g] - S[4096 + g * 64 + i] - S[6144 + g * 64 + i]
                             + yb[(c * NN + i) * NN + i]) * inv2); }
    outb[b * NN + i] = acc;
}

// ------------------------------------------------------------------
// Final combine: action, entropy, a, high
__global__ void k_combine(const float* __restrict__ brA, const float* __restrict__ brB,
                          const float* __restrict__ u, float* __restrict__ out) {
    __shared__ float red[64];
    int b = blockIdx.x, i = threadIdx.x;     // 64 threads
    float a = brA[b * 64 + i] - 2.0f;
    float bma = brB[b * 64 + i] + 4.0f;
    bma = bma > 1e-8f ? bma : 1e-8f;
    out[b * 64 + i]        = a + bma * u[b * 64 + i];  // action (B,1,N)
    out[4160 + b * 64 + i] = a;                        // a (B,N)
    out[8256 + b * 64 + i] = a + bma;                  // high (B,N)
    red[i] = logf(bma);
    __syncthreads();
    for (int s = 32; s > 0; s >>= 1) {
        if (i < s) red[i] += red[i + s];
        __syncthreads();
    }
    if (i == 0) out[4096 + b] = red[0];                // entropy (B,)
}

// ------------------------------------------------------------------
extern "C" void kernel_launch(void* const* d_in, const int* in_sizes, int n_in,
                              void* d_out, int out_size, void* d_ws, size_t ws_size,
                              hipStream_t stream) {
    (void)in_sizes; (void)n_in; (void)out_size; (void)ws_size;
    const float* ch = (const float*)d_in[0];
    const float* u  = (const float*)d_in[1];
    const float* W[2][5]; const float* Bv[2][5];
    for (int s = 0; s < 2; ++s)
        for (int l = 0; l < 5; ++l) {
            W[s][l]  = (const float*)d_in[2 + s * 10 + l * 2];
            Bv[s][l] = (const float*)d_in[2 + s * 10 + l * 2 + 1];
        }

    float* ws = (float*)d_ws;
    float* y  = ws + Y_OFF;
    float* S  = ws + S_OFF;
    float* wt = ws + WT_OFF;
    float* br[2] = { ws + BRA_OFF, ws + BRB_OFF };

    // one-time (per call) weight transpose: [w2a,w3a,w4a,w2b,w3b,w4b]
    k_wt<<<6, 256, 0, stream>>>(W[0][1], W[0][2], W[0][3],
                                W[1][1], W[1][2], W[1][3], wt);

    for (int s = 0; s < 2; ++s) {
        k_conv1<<<NB * 128, 64, 0, stream>>>(ch, W[s][0], Bv[s][0], y);
        for (int l = 1; l <= 3; ++l) {
            k_sums<<<NB * 3, 256, 0, stream>>>(y, S);
            const float* wtl = wt + (size_t)(s * 3 + (l - 1)) * (KPAD * C4);
            k_conv<<<NB * NN, 256, 0, stream>>>(y, ch, S, wtl, Bv[s][l], y);
        }
        k_sums<<<NB * 3, 256, 0, stream>>>(y, S);
        k_diag<<<NB, 64, 0, stream>>>(y, ch, S, W[s][4], Bv[s][4], br[s]);
    }
    k_combine<<<NB, 64, 0, stream>>>(br[0], br[1], u, (float*)d_out);
}